// QGate_65481071396922
// MI455X (gfx1250) — compile-verified
//
#include <hip/hip_runtime.h>
#include <hip/hip_bf16.h>
#include <math.h>

// out[b,i] = prod_{j<=i} cos(params[j]) * cos(x[b,j])   (exact reduction of the
// reference 14-qubit circuit: product state -> CNOT prefix-XOR -> <Z_i>).
// Computed as a log-domain triangular GEMM on the WMMA pipe (f32, 16x16x4),
// signs via prefix-XOR parity of sign bits.

#define NW 14

typedef float v2f __attribute__((ext_vector_type(2)));
typedef float v8f __attribute__((ext_vector_type(8)));

__global__ __launch_bounds__(256) void qexpval_wmma_kernel(
    const float* __restrict__ x,       // [B, 14]
    const float* __restrict__ params,  // [14]
    float* __restrict__ out)           // [B, 14]
{
    // One wave (32 lanes) handles a tile of 16 batch rows.
    __shared__ float    Ls[8][16][16];   // per-wave log-magnitude tile (cols 14,15 = 0)
    __shared__ unsigned Ss[8][16];       // per-row 14-bit sign masks

    const int lane = threadIdx.x & 31;
    const int w    = threadIdx.x >> 5;           // wave in block: 0..7
    const int tile = blockIdx.x * 8 + w;         // 0..31 -> rows tile*16..tile*16+15
    const int hf   = lane >> 4;                  // half-wave select (0/1)
    const int r16  = lane & 15;

    // ---- Phase 1: per-row g = cos(x)*cos(p); stage log|g| and sign mask in LDS
    if (lane < 16) {
        const int row = tile * 16 + lane;
        unsigned sm = 0u;
        #pragma unroll
        for (int j = 0; j < NW; ++j) {
            float g = cosf(x[row * NW + j]) * cosf(params[j]);
            if (g < 0.0f) sm |= (1u << j);
            Ls[w][lane][j] = logf(fmaxf(fabsf(g), 1e-30f));
        }
        Ls[w][lane][14] = 0.0f;
        Ls[w][lane][15] = 0.0f;
        Ss[w][lane] = sm;
    }
    __syncthreads();

    // ---- Phase 2: out_log[M,N] = sum_k L[M,k] * T[k,N],  T[k,N] = (k<=N && k<14)
    // Four chained V_WMMA_F32_16X16X4_F32 cover K = 0..15.
    v8f acc = {};
    #pragma unroll
    for (int q = 0; q < 4; ++q) {
        const int kb = q * 4 + hf * 2;   // this half-wave's K pair: kb, kb+1
        v2f a, b;
        a.x = Ls[w][r16][kb];
        a.y = Ls[w][r16][kb + 1];
        b.x = ((kb     <= r16) && (kb     < NW)) ? 1.0f : 0.0f;
        b.y = ((kb + 1 <= r16) && (kb + 1 < NW)) ? 1.0f : 0.0f;
        acc = __builtin_amdgcn_wmma_f32_16x16x4_f32(
            /*neg_a=*/false, a, /*neg_b=*/false, b,
            /*c_mod=*/(short)0, acc, /*reuse_a=*/false, /*reuse_b=*/false);
    }

    // ---- Phase 3: sign * exp(logsum) -> out.  D layout: M = v + 8*hf, N = r16.
    const int N = r16;
    if (N < NW) {
        const unsigned colmask = (2u << N) - 1u;  // bits 0..N
        #pragma unroll
        for (int v = 0; v < 8; ++v) {
            const int M   = v + hf * 8;
            const int row = tile * 16 + M;
            const unsigned par = __popc(Ss[w][M] & colmask) & 1u;
            const float val = expf(acc[v]);
            out[row * NW + N] = par ? -val : val;
        }
    }
}

extern "C" void kernel_launch(void* const* d_in, const int* in_sizes, int n_in,
                              void* d_out, int out_size, void* d_ws, size_t ws_size,
                              hipStream_t stream) {
    const float* x      = (const float*)d_in[0];   // [512,14] f32
    const float* params = (const float*)d_in[1];   // [14]     f32
    float* out          = (float*)d_out;           // [512,14] f32

    const int nbatch = in_sizes[0] / NW;           // 512
    const int blocks = nbatch / 128;               // 8 tiles of 16 rows per block
    qexpval_wmma_kernel<<<blocks, 256, 0, stream>>>(x, params, out);
}